// GraphConvolution_11390253269508
// MI455X (gfx1250) — compile-verified
//
#include <hip/hip_runtime.h>
#include <hip/hip_bf16.h>

typedef __attribute__((ext_vector_type(2))) float v2f;
typedef __attribute__((ext_vector_type(8))) float v8f;

#define DIN  256
#define DOUT 256

// ---------------------------------------------------------------------------
// Kernel 1: support = X @ W using V_WMMA_F32_16X16X4_F32 (CDNA5 wave32 WMMA).
// Block = 256 threads = 8 waves. Wave w handles M-tile (blockIdx.x*8 + w)
// (16 rows) and a 64-wide N strip (blockIdx.y), i.e. four 16x16 C tiles,
// reusing the A operand across the 4 WMMAs per K-step.
//
// A (16x4 f32) lane layout: lanes 0-15 hold rows M=0..15 with K={0,1} in
// VGPR0/VGPR1; lanes 16-31 hold the same rows with K={2,3}.
// B (4x16 f32): lane l<16 holds N=l with K={0,1}; lanes 16-31 hold K={2,3}.
// C/D (16x16 f32): VGPR r = row M=r (lanes 0-15) / M=8+r (lanes 16-31).
// ---------------------------------------------------------------------------
__global__ __launch_bounds__(256)
void gcn_gemm_wmma(const float* __restrict__ X, const float* __restrict__ W,
                   float* __restrict__ S, int n_rows) {
  const int wave  = threadIdx.x >> 5;
  const int lane  = threadIdx.x & 31;
  const int mtile = blockIdx.x * 8 + wave;
  if (mtile * 16 >= n_rows) return;

  const int m0    = mtile * 16;
  const int n0    = blockIdx.y * 64;
  const int lrow  = lane & 15;          // row (A) / column (B) within tile
  const int khalf = (lane >> 4) << 1;   // 0 for lanes 0-15, 2 for lanes 16-31

  v8f acc0 = {}; v8f acc1 = {}; v8f acc2 = {}; v8f acc3 = {};

  const float* xrow = X + (size_t)(m0 + lrow) * DIN;

  for (int k = 0; k < DIN; k += 4) {
    // A operand: two consecutive K values for this lane's row (8B-aligned).
    v2f a = *(const v2f*)(xrow + k + khalf);

    // B operand: W[k+khalf][n], W[k+khalf+1][n] for the 4 N sub-tiles.
    const float* wbase = W + (size_t)(k + khalf) * DOUT + n0 + lrow;
    v2f b0, b1, b2, b3;
    b0.x = wbase[0];   b0.y = wbase[DOUT + 0];
    b1.x = wbase[16];  b1.y = wbase[DOUT + 16];
    b2.x = wbase[32];  b2.y = wbase[DOUT + 32];
    b3.x = wbase[48];  b3.y = wbase[DOUT + 48];

    acc0 = __builtin_amdgcn_wmma_f32_16x16x4_f32(false, a, false, b0,
                                                 (short)0, acc0, false, false);
    acc1 = __builtin_amdgcn_wmma_f32_16x16x4_f32(false, a, false, b1,
                                                 (short)0, acc1, false, false);
    acc2 = __builtin_amdgcn_wmma_f32_16x16x4_f32(false, a, false, b2,
                                                 (short)0, acc2, false, false);
    acc3 = __builtin_amdgcn_wmma_f32_16x16x4_f32(false, a, false, b3,
                                                 (short)0, acc3, false, false);
  }

  // Store C tiles: VGPR r -> row rbase + r, this lane's column.
  const int rbase = m0 + ((lane >> 4) << 3);
  const int cbase = n0 + lrow;
  #pragma unroll
  for (int r = 0; r < 8; ++r) {
    float* orow = S + (size_t)(rbase + r) * DOUT + cbase;
    orow[0]  = acc0[r];
    orow[16] = acc1[r];
    orow[32] = acc2[r];
    orow[48] = acc3[r];
  }
}

// ---------------------------------------------------------------------------
// Kernel 2: out[row][d] = bias[d]  (d_out is poisoned by the harness).
// One block per output row, 256 threads = 256 output columns.
// ---------------------------------------------------------------------------
__global__ __launch_bounds__(256)
void gcn_bias_init(const float* __restrict__ bias, float* __restrict__ out) {
  out[(size_t)blockIdx.x * DOUT + threadIdx.x] = bias[threadIdx.x];
}

// ---------------------------------------------------------------------------
// Kernel 3: edge scatter-add. One wave per edge:
//   out[row[e]][:] += val[e] * support[col[e]][:]
// Each lane: two float4 gathers (b128 loads) + 8 hardware f32 atomics
// (unsafeAtomicAdd -> global_atomic_add_f32, no CAS loop).
// ---------------------------------------------------------------------------
__global__ __launch_bounds__(256)
void gcn_spmm_scatter(const int* __restrict__ adj_row,
                      const int* __restrict__ adj_col,
                      const float* __restrict__ adj_val,
                      const float* __restrict__ S,
                      float* __restrict__ out, int n_edges) {
  const int wid  = (int)((blockIdx.x * (unsigned)blockDim.x + threadIdx.x) >> 5);
  const int lane = threadIdx.x & 31;
  if (wid >= n_edges) return;

  const int   r = adj_row[wid];
  const int   c = adj_col[wid];
  const float v = adj_val[wid];

  const float4* src = (const float4*)(S + (size_t)c * DOUT);
  float*        dst = out + (size_t)r * DOUT;

  float4 g0 = src[lane];        // cols [lane*4, lane*4+3]
  float4 g1 = src[lane + 32];   // cols [128 + lane*4, ...]

  const int c0 = lane * 4;
  unsafeAtomicAdd(dst + c0 + 0,       g0.x * v);
  unsafeAtomicAdd(dst + c0 + 1,       g0.y * v);
  unsafeAtomicAdd(dst + c0 + 2,       g0.z * v);
  unsafeAtomicAdd(dst + c0 + 3,       g0.w * v);
  unsafeAtomicAdd(dst + c0 + 128 + 0, g1.x * v);
  unsafeAtomicAdd(dst + c0 + 128 + 1, g1.y * v);
  unsafeAtomicAdd(dst + c0 + 128 + 2, g1.z * v);
  unsafeAtomicAdd(dst + c0 + 128 + 3, g1.w * v);
}

// ---------------------------------------------------------------------------
extern "C" void kernel_launch(void* const* d_in, const int* in_sizes, int n_in,
                              void* d_out, int out_size, void* d_ws, size_t ws_size,
                              hipStream_t stream) {
  const float* X       = (const float*)d_in[0];  // [N, 256]
  const int*   adj_row = (const int*)  d_in[1];  // [E]
  const int*   adj_col = (const int*)  d_in[2];  // [E]
  const float* adj_val = (const float*)d_in[3];  // [E]
  const float* W       = (const float*)d_in[4];  // [256, 256]
  const float* bias    = (const float*)d_in[5];  // [256]
  float*       out     = (float*)d_out;          // [N, 256]
  float*       support = (float*)d_ws;           // [N, 256] scratch (102.4 MB)

  const int n_rows  = in_sizes[0] / DIN;
  const int n_edges = in_sizes[1];

  // 1) support = X @ W   (WMMA f32)
  const int mtiles = (n_rows + 15) / 16;
  dim3 g1((mtiles + 7) / 8, DOUT / 64);
  gcn_gemm_wmma<<<g1, 256, 0, stream>>>(X, W, support, n_rows);

  // 2) out = bias (broadcast)
  gcn_bias_init<<<n_rows, DOUT, 0, stream>>>(bias, out);

  // 3) out[row] += val * support[col]   (one wave per edge)
  const int waves_per_block = 256 / 32;
  const int nblocks = (n_edges + waves_per_block - 1) / waves_per_block;
  gcn_spmm_scatter<<<nblocks, 256, 0, stream>>>(adj_row, adj_col, adj_val,
                                                support, out, n_edges);
}